// ContinuousDeepFM_53901839565364
// MI455X (gfx1250) — compile-verified
//
#include <hip/hip_runtime.h>
#include <hip/hip_bf16.h>

typedef __attribute__((ext_vector_type(2))) float v2f;
typedef __attribute__((ext_vector_type(8))) float v8f;

#define DIM 512

// ---------------------------------------------------------------------------
// fp32 WMMA GEMM:  C[M,N] = op(A[M,K] x B) (+ bias) (+ relu)
//   TRANS_B = false : B is row-major [K,N]
//   TRANS_B = true  : B is W row-major [N,K], logical B = W^T
//
// Block: 256 threads = 8 waves arranged 4(M) x 2(N).
// Wave tile: 32(M) x 64(N) = 2 M-subtiles x 4 N-subtiles -> 8 accumulators.
// Per K-step (K+=4): 2 A-fragment loads + 4 B-fragment loads feed 8 WMMAs,
// software-pipelined (next K-step's fragments prefetched before the WMMAs).
// Grid: (N/128, M/128).
// ---------------------------------------------------------------------------
template <bool TRANS_B, bool RELU, bool HAS_BIAS>
__global__ __launch_bounds__(256) void wmma_gemm_f32(
    const float* __restrict__ A, const float* __restrict__ Bm,
    const float* __restrict__ bias, float* __restrict__ C,
    int M, int N, int K) {
  const int lane  = threadIdx.x & 31;
  const int wave  = threadIdx.x >> 5;
  const int waveM = wave >> 1;   // 0..3
  const int waveN = wave & 1;    // 0..1
  const int m16   = lane & 15;   // M index within A tile / N index within B tile
  const int khalf = lane >> 4;   // 0: K={0,1}, 1: K={2,3}

  const int tileM = blockIdx.y * 128 + waveM * 32;
  const int tileN = blockIdx.x * 128 + waveN * 64;

  v8f acc[2][4] = {{v8f{}, v8f{}, v8f{}, v8f{}}, {v8f{}, v8f{}, v8f{}, v8f{}}};

  // Per-lane base pointers (K-offset folds into immediate offsets).
  const float* Arow0 = A + (size_t)(tileM + m16) * K + khalf * 2;
  const float* Arow1 = Arow0 + (size_t)16 * K;
  const float* Bp[4];
#pragma unroll
  for (int t = 0; t < 4; ++t) {
    const int n = tileN + t * 16 + m16;
    if (TRANS_B) {
      Bp[t] = Bm + (size_t)n * K + khalf * 2;  // contiguous float2 along K
    } else {
      Bp[t] = Bm + (size_t)(khalf * 2) * N + n;  // two strided b32 loads
    }
  }

  auto loadB = [&](int k, v2f* b) {
#pragma unroll
    for (int t = 0; t < 4; ++t) {
      if (TRANS_B) {
        b[t] = *(const v2f*)(Bp[t] + k);
      } else {
        b[t].x = Bp[t][(size_t)k * N];
        b[t].y = Bp[t][(size_t)(k + 1) * N];
      }
    }
  };

  // Prologue: fragments for k = 0.
  v2f a0 = *(const v2f*)(Arow0);
  v2f a1 = *(const v2f*)(Arow1);
  v2f b[4];
  loadB(0, b);

  for (int k = 0; k < K - 4; k += 4) {
    // Prefetch next K-step's fragments (independent of the WMMAs below).
    const int kn = k + 4;
    v2f a0n = *(const v2f*)(Arow0 + kn);
    v2f a1n = *(const v2f*)(Arow1 + kn);
    v2f bn[4];
    loadB(kn, bn);

    // 8 independent WMMAs on the current fragments.
#pragma unroll
    for (int t = 0; t < 4; ++t)
      acc[0][t] = __builtin_amdgcn_wmma_f32_16x16x4_f32(
          false, a0, false, b[t], (short)0, acc[0][t], false, false);
#pragma unroll
    for (int t = 0; t < 4; ++t)
      acc[1][t] = __builtin_amdgcn_wmma_f32_16x16x4_f32(
          false, a1, false, b[t], (short)0, acc[1][t], false, false);

    a0 = a0n;
    a1 = a1n;
#pragma unroll
    for (int t = 0; t < 4; ++t) b[t] = bn[t];
  }

  // Epilogue K-step.
#pragma unroll
  for (int t = 0; t < 4; ++t)
    acc[0][t] = __builtin_amdgcn_wmma_f32_16x16x4_f32(
        false, a0, false, b[t], (short)0, acc[0][t], false, false);
#pragma unroll
  for (int t = 0; t < 4; ++t)
    acc[1][t] = __builtin_amdgcn_wmma_f32_16x16x4_f32(
        false, a1, false, b[t], (short)0, acc[1][t], false, false);

  // Store: C/D layout -> VGPR i holds row (i + 8*khalf), col = m16 (per tile).
#pragma unroll
  for (int ms = 0; ms < 2; ++ms) {
    const int rowBase = tileM + ms * 16 + 8 * khalf;
#pragma unroll
    for (int t = 0; t < 4; ++t) {
      const int n = tileN + t * 16 + m16;
      const float bv = HAS_BIAS ? bias[n] : 0.0f;
#pragma unroll
      for (int i = 0; i < 8; ++i) {
        float v = acc[ms][t][i] + bv;
        if (RELU) v = fmaxf(v, 0.0f);
        C[(size_t)(rowBase + i) * N + n] = v;
      }
    }
  }
}

// ---------------------------------------------------------------------------
// Fused FM second-order + final sum.
// second_order[b,j] = 0.5*(S2[b] - S1[b]^2) * xw[b,j]^2,
//   with S1[b]=sum_i x[b,i], S2[b]=sum_i x[b,i]^2.
// out = first_order + second_order + deep_out.   One block per batch row.
// ---------------------------------------------------------------------------
__global__ __launch_bounds__(256) void deepfm_combine(
    const float* __restrict__ x, const float* __restrict__ fo,
    const float* __restrict__ xw, const float* __restrict__ h,
    float* __restrict__ out, int D) {
  const int b = blockIdx.x;
  __shared__ float sS1[8];
  __shared__ float sS2[8];

  float s1 = 0.0f, s2 = 0.0f;
  for (int j = threadIdx.x; j < D; j += blockDim.x) {
    float v = x[(size_t)b * D + j];
    s1 += v;
    s2 += v * v;
  }
#pragma unroll
  for (int off = 16; off > 0; off >>= 1) {
    s1 += __shfl_down(s1, off, 32);
    s2 += __shfl_down(s2, off, 32);
  }
  const int lane = threadIdx.x & 31;
  const int wave = threadIdx.x >> 5;
  if (lane == 0) { sS1[wave] = s1; sS2[wave] = s2; }
  __syncthreads();

  float S1 = 0.0f, S2 = 0.0f;
#pragma unroll
  for (int w = 0; w < 8; ++w) { S1 += sS1[w]; S2 += sS2[w]; }
  const float coef = 0.5f * (S2 - S1 * S1);

  for (int j = threadIdx.x; j < D; j += blockDim.x) {
    const size_t idx = (size_t)b * D + j;
    const float w = xw[idx];
    out[idx] = fo[idx] + coef * (w * w) + h[idx];
  }
}

extern "C" void kernel_launch(void* const* d_in, const int* in_sizes, int n_in,
                              void* d_out, int out_size, void* d_ws, size_t ws_size,
                              hipStream_t stream) {
  (void)in_sizes; (void)n_in; (void)out_size; (void)ws_size;
  const int D = DIM, B = DIM;  // 512 x 512

  const float* x     = (const float*)d_in[0];
  const float* w1    = (const float*)d_in[1];  // first_order_weights [D,D]
  const float* bias  = (const float*)d_in[2];  // [D]
  const float* w2    = (const float*)d_in[3];  // second_order_weights [D,D]
  const float* wf    = (const float*)d_in[4];  // feature_weights [D,D]
  const float* mlp_w = (const float*)d_in[5];  // [4,D,D]
  const float* mlp_b = (const float*)d_in[6];  // [4,D]
  float* out = (float*)d_out;

  float* ws = (float*)d_ws;
  const size_t mat = (size_t)B * D;
  float* fo = ws;            // first_order
  float* xw = ws + mat;      // x @ W2
  float* h0 = ws + 2 * mat;  // deep ping
  float* h1 = ws + 3 * mat;  // deep pong

  dim3 grid(D / 128, B / 128);
  dim3 block(256);

  // first_order = x @ W1 + bias
  wmma_gemm_f32<false, false, true><<<grid, block, 0, stream>>>(x, w1, bias, fo, B, D, D);
  // xw = x @ W2
  wmma_gemm_f32<false, false, false><<<grid, block, 0, stream>>>(x, w2, nullptr, xw, B, D, D);
  // h = x @ Wf
  wmma_gemm_f32<false, false, false><<<grid, block, 0, stream>>>(x, wf, nullptr, h0, B, D, D);
  // 3 hidden layers: h = relu(h @ mlp_w[i]^T + mlp_b[i])
  wmma_gemm_f32<true, true, true><<<grid, block, 0, stream>>>(h0, mlp_w + 0 * mat, mlp_b + 0 * D, h1, B, D, D);
  wmma_gemm_f32<true, true, true><<<grid, block, 0, stream>>>(h1, mlp_w + 1 * mat, mlp_b + 1 * D, h0, B, D, D);
  wmma_gemm_f32<true, true, true><<<grid, block, 0, stream>>>(h0, mlp_w + 2 * mat, mlp_b + 2 * D, h1, B, D, D);
  // output layer (no relu)
  wmma_gemm_f32<true, false, true><<<grid, block, 0, stream>>>(h1, mlp_w + 3 * mat, mlp_b + 3 * D, h0, B, D, D);

  // out = first_order + 0.5*(S2 - S1^2)*xw^2 + deep
  deepfm_combine<<<dim3(B), block, 0, stream>>>(x, fo, xw, h0, out, D);
}